// QuantumAttnBlk_6760278524575
// MI455X (gfx1250) — compile-verified
//
#include <hip/hip_runtime.h>
#include <hip/hip_bf16.h>

typedef __attribute__((ext_vector_type(2))) float v2f;
typedef __attribute__((ext_vector_type(8))) float v8f;

#define PSI_STRIDE 34   // 32 floats per sample + 2 pad (keeps b64 alignment, spreads banks)

// ds_swizzle group-of-32 immediates: offset = (xor<<10) | (or<<5) | and(0x1f)
#define SWZ_XOR1 0x041F
#define SWZ_XOR2 0x081F
#define SWZ_XOR4 0x101F
#define SWZ_XOR8 0x201F

// ---------------------------------------------------------------------------
// prep_kernel: build the 32x32 real matrix W encoding phi = (kronU3 o CNOTperm) psi
// and store it in WMMA B-fragment order into d_ws (1024 floats).
// Fragment order: wfrag[((ntile*8 + kk)*2 + v)*32 + lane]
//   = W[ kk*4 + v + 2*(lane/16) ][ ntile*16 + (lane%16) ]
// ---------------------------------------------------------------------------
__global__ __launch_bounds__(32) void qsim_prep_kernel(const float* __restrict__ weights,
                                                       float* __restrict__ wfrag) {
    if (threadIdx.x != 0) return;

    // Per-wire U3 matrices: [[c, -e^{il}s],[e^{ip}s, e^{i(p+l)}c]]
    float ur[4][2][2], ui[4][2][2];
    for (int w = 0; w < 4; ++w) {
        float t = weights[w * 3 + 0];
        float p = weights[w * 3 + 1];
        float l = weights[w * 3 + 2];
        float c, s, cp, sp, cl, sl;
        __sincosf(0.5f * t, &s, &c);
        __sincosf(p, &sp, &cp);
        __sincosf(l, &sl, &cl);
        ur[w][0][0] = c;       ui[w][0][0] = 0.0f;
        ur[w][0][1] = -cl * s; ui[w][0][1] = -sl * s;
        ur[w][1][0] = cp * s;  ui[w][1][0] = sp * s;
        float cpl = cp * cl - sp * sl;
        float spl = sp * cl + cp * sl;
        ur[w][1][1] = cpl * c; ui[w][1][1] = spl * c;
    }

    // CNOT chain permutation: psi4_b = psi_{g(b)}, g = f1(f2(f3(f4(b))))
    // wire w <-> bit (3-w).  CNOT(c,t): if bit(3-c) set, flip bit(3-t).
    int g[16];
    for (int b = 0; b < 16; ++b) {
        int idx = b;
        if (idx & 1) idx ^= 8;  // f4 = CNOT(3,0)
        if (idx & 2) idx ^= 1;  // f3 = CNOT(2,3)
        if (idx & 4) idx ^= 2;  // f2 = CNOT(1,2)
        if (idx & 8) idx ^= 4;  // f1 = CNOT(0,1)
        g[b] = idx;
    }

    // M[s][g(t)] = K[s][t] where K = U3_0 (x) U3_1 (x) U3_2 (x) U3_3
    float Mre[16][16], Mim[16][16];
    for (int s = 0; s < 16; ++s) {
        for (int t = 0; t < 16; ++t) {
            float kr = 1.0f, ki = 0.0f;
            for (int w = 0; w < 4; ++w) {
                int sw = (s >> (3 - w)) & 1;
                int tw = (t >> (3 - w)) & 1;
                float er = ur[w][sw][tw], ei = ui[w][sw][tw];
                float nr = kr * er - ki * ei;
                float ni = kr * ei + ki * er;
                kr = nr; ki = ni;
            }
            Mre[s][g[t]] = kr;
            Mim[s][g[t]] = ki;
        }
    }

    // Real 32x32 W:  [RePhi | ImPhi] = [RePsi | ImPsi] * W
    float W[32][32];
    for (int s = 0; s < 16; ++s) {
        for (int u = 0; u < 16; ++u) {
            W[u][s]           =  Mre[s][u];
            W[16 + u][s]      = -Mim[s][u];
            W[u][16 + s]      =  Mim[s][u];
            W[16 + u][16 + s] =  Mre[s][u];
        }
    }

    // Store in B-fragment order
    for (int nt = 0; nt < 2; ++nt)
        for (int kk = 0; kk < 8; ++kk)
            for (int v = 0; v < 2; ++v)
                for (int lane = 0; lane < 32; ++lane) {
                    int K = kk * 4 + v + 2 * (lane / 16);
                    int N = nt * 16 + (lane % 16);
                    wfrag[((nt * 8 + kk) * 2 + v) * 32 + lane] = W[K][N];
                }
}

// ---------------------------------------------------------------------------
// Main kernel: 256 threads = 8 wave32 per block, 256 samples per block.
// Phase 1: each thread builds its sample's product state (32 reals) into LDS.
// Phase 2: each wave runs two 16-sample WMMA tiles of Psi(16x32) x W(32x32),
//          squares Re/Im, and does a 4-stage Walsh-Hadamard butterfly with
//          ds_swizzle to extract the four <Z_q> values per row.
// Phase 3: coalesced float4 store of the staged outputs.
// ---------------------------------------------------------------------------
__global__ __launch_bounds__(256) void qsim_kernel(const float* __restrict__ x,
                                                   const float* __restrict__ wfrag,
                                                   float* __restrict__ out) {
    __shared__ float lds[256 * PSI_STRIDE];
    __shared__ float outstage[256 * 4];

    const int tid = threadIdx.x;
    const long long sample0 = (long long)blockIdx.x * 256;

    // ---------------- Phase 1: product state ----------------
    {
        const float* xp = x + (sample0 + tid) * 12;
        float ar[4], ai[4], br[4], bi[4];
        const float k = 0.70710678118654752f;
#pragma unroll
        for (int w = 0; w < 4; ++w) {
            float c0, s0, c1, s1, c2, s2;
            __sincosf(xp[3 * w + 0] * 0.5f, &s0, &c0);
            __sincosf(xp[3 * w + 1] * 0.5f, &s1, &c1);
            __sincosf(xp[3 * w + 2] * 0.5f, &s2, &c2);
            // after H, RZ(t0), RY(t1):
            float are = k * (c1 - s1) * c0, aim = -k * (c1 + s1) * s0;
            float bre = k * (c1 + s1) * c0, bim =  k * (c1 - s1) * s0;
            // RZ(t2): a *= (c2 - i s2), b *= (c2 + i s2)
            ar[w] = are * c2 + aim * s2;  ai[w] = aim * c2 - are * s2;
            br[w] = bre * c2 - bim * s2;  bi[w] = bim * c2 + bre * s2;
        }
        float q01r[4], q01i[4], q23r[4], q23i[4];
#pragma unroll
        for (int i = 0; i < 2; ++i)
#pragma unroll
            for (int j = 0; j < 2; ++j) {
                float xr = i ? br[0] : ar[0], xi = i ? bi[0] : ai[0];
                float yr = j ? br[1] : ar[1], yi = j ? bi[1] : ai[1];
                q01r[i * 2 + j] = xr * yr - xi * yi;
                q01i[i * 2 + j] = xr * yi + xi * yr;
                float zr = i ? br[2] : ar[2], zi = i ? bi[2] : ai[2];
                float wr = j ? br[3] : ar[3], wi = j ? bi[3] : ai[3];
                q23r[i * 2 + j] = zr * wr - zi * wi;
                q23i[i * 2 + j] = zr * wi + zi * wr;
            }
        float* row = &lds[tid * PSI_STRIDE];
#pragma unroll
        for (int i = 0; i < 4; ++i)
#pragma unroll
            for (int j = 0; j < 4; ++j) {
                float rr = q01r[i] * q23r[j] - q01i[i] * q23i[j];
                float ii = q01r[i] * q23i[j] + q01i[i] * q23r[j];
                row[i * 4 + j]      = rr;   // Re psi, k = 0..15
                row[16 + i * 4 + j] = ii;   // Im psi, k = 16..31
            }
    }

    // Load B fragments (uniform, 4 KB, lives in L2) before the barrier.
    const int lane = tid & 31;
    const int wave = tid >> 5;
    v2f bfrag[2][8];
#pragma unroll
    for (int nt = 0; nt < 2; ++nt)
#pragma unroll
        for (int kk = 0; kk < 8; ++kk) {
            bfrag[nt][kk].x = wfrag[((nt * 8 + kk) * 2 + 0) * 32 + lane];
            bfrag[nt][kk].y = wfrag[((nt * 8 + kk) * 2 + 1) * 32 + lane];
        }

    // WHT butterfly per-lane signs via bit ops: +-1.0f with sign = lane bit m
    const unsigned ul = (unsigned)lane;
    const float sg1 = __int_as_float(0x3f800000u | ((ul & 1u) << 31));
    const float sg2 = __int_as_float(0x3f800000u | ((ul & 2u) << 30));
    const float sg4 = __int_as_float(0x3f800000u | ((ul & 4u) << 29));
    const float sg8 = __int_as_float(0x3f800000u | ((ul & 8u) << 28));

    // Which lanes hold final <Z_q>: h = lane&15 in {8,4,2,1} -> q = 0..3
    const int h = lane & 15;
    const bool active = (h == 1) || (h == 2) || (h == 4) || (h == 8);
    const int qidx = (h == 8) ? 0 : (h == 4) ? 1 : (h == 2) ? 2 : 3;
    const int rowoff = (lane >> 4) << 3;   // upper half handles rows r+8

    __syncthreads();

    // ---------------- Phase 2: WMMA entangle + measure ----------------
    const int mrow = lane & 15;          // A-matrix M index
    const int koff = (lane >> 4) * 2;    // A-matrix K sub-offset per ISA layout

#pragma unroll
    for (int t = 0; t < 2; ++t) {
        const int tileBase = wave * 32 + t * 16;
        const float* arow = &lds[(tileBase + mrow) * PSI_STRIDE];

        v8f acc0 = {};  // Re(phi), columns 0..15
        v8f acc1 = {};  // Im(phi), columns 0..15
#pragma unroll
        for (int kk = 0; kk < 8; ++kk) {
            v2f a;
            a.x = arow[kk * 4 + koff + 0];
            a.y = arow[kk * 4 + koff + 1];
            acc0 = __builtin_amdgcn_wmma_f32_16x16x4_f32(
                false, a, false, bfrag[0][kk], (short)0, acc0, false, false);
            acc1 = __builtin_amdgcn_wmma_f32_16x16x4_f32(
                false, a, false, bfrag[1][kk], (short)0, acc1, false, false);
        }

        v8f p = acc0 * acc0 + acc1 * acc1;   // probabilities, C/D layout

        // 4-stage Walsh-Hadamard butterfly per row; lane h=1<<(3-q) ends with
        // sum_n (-1)^{bit(3-q) of n} p_n, i.e. <Z_q>.
#pragma unroll
        for (int r = 0; r < 8; ++r) {
            float v = p[r];
            float tt;
            tt = __int_as_float(__builtin_amdgcn_ds_swizzle(__float_as_int(v), SWZ_XOR1));
            v  = fmaf(sg1, v, tt);
            tt = __int_as_float(__builtin_amdgcn_ds_swizzle(__float_as_int(v), SWZ_XOR2));
            v  = fmaf(sg2, v, tt);
            tt = __int_as_float(__builtin_amdgcn_ds_swizzle(__float_as_int(v), SWZ_XOR4));
            v  = fmaf(sg4, v, tt);
            tt = __int_as_float(__builtin_amdgcn_ds_swizzle(__float_as_int(v), SWZ_XOR8));
            v  = fmaf(sg8, v, tt);
            if (active)
                outstage[(tileBase + r + rowoff) * 4 + qidx] = v;
        }
    }

    // ---------------- Phase 3: coalesced output ----------------
    __syncthreads();
    float4 o = ((const float4*)outstage)[tid];
    ((float4*)out)[sample0 + tid] = o;
}

// ---------------------------------------------------------------------------
extern "C" void kernel_launch(void* const* d_in, const int* in_sizes, int n_in,
                              void* d_out, int out_size, void* d_ws, size_t ws_size,
                              hipStream_t stream) {
    const float* x       = (const float*)d_in[0];   // (B, 12) f32
    const float* weights = (const float*)d_in[1];   // (1, 4, 3) f32
    float* out   = (float*)d_out;                   // (B, 4) f32
    float* wfrag = (float*)d_ws;                    // 1024 f32 = 4 KB

    qsim_prep_kernel<<<1, 32, 0, stream>>>(weights, wfrag);

    const int B = in_sizes[0] / 12;                 // 524288
    qsim_kernel<<<B / 256, 256, 0, stream>>>(x, wfrag, out);
}